// Sparse_2954937500105
// MI455X (gfx1250) — compile-verified
//
#include <hip/hip_runtime.h>

#define NCOLS 8192   // input feature dim
#define NROWS 8192   // output feature dim
#define BATCH 1024   // batch
#define KNNZ  32     // nnz per output row
#define RT    16     // output rows per block (== TDM gather indices per descriptor)

typedef __attribute__((ext_vector_type(4))) unsigned int v4u;
typedef __attribute__((ext_vector_type(8))) int          v8i;
typedef __attribute__((ext_vector_type(4))) int          v4i;

// ---------------------------------------------------------------------------
// Main kernel (placed first so the disasm snippet shows the TDM path):
// per block: 16 output rows x full batch (1024).
// For each k in [0,32): one TDM gather-mode descriptor (16x 16-bit row
// indices) fetches the 16 needed input columns (16 x 4KB contiguous runs)
// into a double-buffered LDS stage; 8 waves FMA-accumulate from LDS.
// ---------------------------------------------------------------------------
__global__ __launch_bounds__(256) void spmm_gather(const float* __restrict__ inT,
                                                   const float* __restrict__ vals,
                                                   const int*   __restrict__ cols,
                                                   float* __restrict__ out) {
    __shared__ float buf[2][RT * BATCH];   // 2 x 64KB double-buffered gather stage
    __shared__ float sval[RT][KNNZ];       // 2KB weights
    __shared__ int   scol[RT][KNNZ];       // 2KB column indices

    const int r0 = blockIdx.x * RT;
    const int t  = threadIdx.x;

    // Preload this block's (col, val) metadata.
    for (int i = t; i < RT * KNNZ; i += 256) {
        ((int*)scol)[i]   = cols[(size_t)r0 * KNNZ + i];
        ((float*)sval)[i] = vals[(size_t)r0 * KNNZ + i];
    }
    __syncthreads();

    // ---- Loop-invariant descriptor halves (built once) --------------------
    // Group 0: count=1, gather_mode=1 (16-bit indices), lds_addr, global base,
    //          type=2 ("image").  Two variants, one per LDS buffer.
    const unsigned long long ga = (unsigned long long)inT;
    v4u g0buf[2];
#pragma unroll
    for (int s = 0; s < 2; ++s) {
        g0buf[s][0] = 1u | (1u << 31);
        g0buf[s][1] = (unsigned)(unsigned long long)(&buf[s][0]);   // lds_addr
        g0buf[s][2] = (unsigned)(ga & 0xffffffffull);               // global_addr lo
        g0buf[s][3] = (unsigned)((ga >> 32) & 0x01ffffffull) | (2u << 30);
    }
    // Group 1: data_size=4B, tensor_dim0=BATCH, tensor_dim1=NCOLS,
    //          tile_dim0=BATCH, tile_dim1=RT indices, dim0 stride=BATCH.
    v8i g1;
    g1[0] = (2 << 16);
    g1[1] = (int)((BATCH & 0xffff) << 16);
    g1[2] = (BATCH >> 16) | ((NCOLS & 0xffff) << 16);
    g1[3] = (NCOLS >> 16) | (BATCH << 16);
    g1[4] = RT;
    g1[5] = BATCH;
    g1[6] = 0;
    g1[7] = 0;
    const v8i g4 = {0, 0, 0, 0, 0, 0, 0, 0};   // unused trailing group (clang-23 form)

    // Per-step variable part: groups 2/3 = 16 packed 16-bit gather indices.
    auto issue = [&](int k, int sel) {
        v4i g2, g3;
#pragma unroll
        for (int p = 0; p < 4; ++p) {
            g2[p] = (scol[2 * p + 0][k] & 0xffff) | (scol[2 * p + 1][k] << 16);
            g3[p] = (scol[2 * p + 8][k] & 0xffff) | (scol[2 * p + 9][k] << 16);
        }
        __builtin_amdgcn_tensor_load_to_lds(g0buf[sel], g1, g2, g3, g4, 0);
    };

    float acc[RT][4];
#pragma unroll
    for (int r = 0; r < RT; ++r)
        acc[r][0] = acc[r][1] = acc[r][2] = acc[r][3] = 0.0f;

    if (t < 32) issue(0, 0);                           // wave 0 drives the TDM

    for (int k = 0; k < KNNZ; ++k) {
        __syncthreads();                               // buffer (k+1)&1 free for reuse
        if (t < 32) {
            if (k + 1 < KNNZ) {
                issue(k + 1, (k + 1) & 1);
                __builtin_amdgcn_s_wait_tensorcnt(1);  // step k landed, k+1 in flight
            } else {
                __builtin_amdgcn_s_wait_tensorcnt(0);
            }
        }
        __syncthreads();                               // gathered data visible to all waves

        const float4* bv = (const float4*)&buf[k & 1][0];
#pragma unroll
        for (int r = 0; r < RT; ++r) {
            const float4 x = bv[r * (BATCH / 4) + t];  // batch 4t..4t+3 of column r
            const float  w = sval[r][k];
            acc[r][0] = fmaf(w, x.x, acc[r][0]);
            acc[r][1] = fmaf(w, x.y, acc[r][1]);
            acc[r][2] = fmaf(w, x.z, acc[r][2]);
            acc[r][3] = fmaf(w, x.w, acc[r][3]);
        }
    }

    // Store: thread t owns batch rows b = 4t..4t+3; each writes 16 consecutive
    // output features as 4 x float4 (64B contiguous per batch row).
#pragma unroll
    for (int j = 0; j < 4; ++j) {
        float* op = out + (size_t)(4 * t + j) * NROWS + r0;
#pragma unroll
        for (int rr = 0; rr < 4; ++rr) {
            float4 v;
            v.x = acc[4 * rr + 0][j];
            v.y = acc[4 * rr + 1][j];
            v.z = acc[4 * rr + 2][j];
            v.w = acc[4 * rr + 3][j];
            *(float4*)(op + 4 * rr) = v;
        }
    }
}

// ---------------------------------------------------------------------------
// Kernel 1: transpose inputs (BATCH x NCOLS) -> inT (NCOLS x BATCH)
// so that gathered "columns" become contiguous 4KB runs (full cachelines).
// ---------------------------------------------------------------------------
__global__ __launch_bounds__(256) void transpose_in(const float* __restrict__ in,
                                                    float* __restrict__ oT) {
    __shared__ float tile[32][33];
    const int cx = blockIdx.x * 32 + threadIdx.x;   // column index in `in`
    const int ry = blockIdx.y * 32;                 // row (batch) base in `in`
#pragma unroll
    for (int i = threadIdx.y; i < 32; i += 8)
        tile[i][threadIdx.x] = in[(size_t)(ry + i) * NCOLS + cx];
    __syncthreads();
    const int bx = blockIdx.y * 32 + threadIdx.x;   // batch index in oT
    const int cy = blockIdx.x * 32;                 // column base in oT
#pragma unroll
    for (int i = threadIdx.y; i < 32; i += 8)
        oT[(size_t)(cy + i) * BATCH + bx] = tile[threadIdx.x][i];
}

// ---------------------------------------------------------------------------
extern "C" void kernel_launch(void* const* d_in, const int* in_sizes, int n_in,
                              void* d_out, int out_size, void* d_ws, size_t ws_size,
                              hipStream_t stream) {
    const float* inputs = (const float*)d_in[0];   // (B, NCOLS) f32
    const float* skern  = (const float*)d_in[1];   // (NNZ, 1)  f32
    // d_in[2] = rows (implied: r = k / 32, sorted & dense) -- unused
    const int*   cols   = (const int*)d_in[3];     // (NNZ,) int
    float*       out    = (float*)d_out;           // (B, NROWS) f32
    float*       inT    = (float*)d_ws;            // 32MB transposed inputs

    dim3 bA(32, 8), gA(NCOLS / 32, BATCH / 32);
    transpose_in<<<gA, bA, 0, stream>>>(inputs, inT);

    spmm_gather<<<NROWS / RT, 256, 0, stream>>>(inT, skern, cols, out);
}